// MultiheadAttention_85624468013408
// MI455X (gfx1250) — compile-verified
//
#include <hip/hip_runtime.h>
#include <hip/hip_bf16.h>

// Problem constants (from reference)
static constexpr int B_SZ   = 128;
static constexpr int LEN    = 64;
static constexpr int C_GRP  = 8;
static constexpr int NHEAD  = 8;
static constexpr int DK     = 64;
static constexpr int D_IN   = 512;   // DM * C
static constexpr int D_PROJ = 4096;  // NH * DK * C

typedef __attribute__((ext_vector_type(2))) float v2f;
typedef __attribute__((ext_vector_type(8))) float v8f;
typedef __attribute__((ext_vector_type(4))) int   v4i;

typedef __attribute__((address_space(1))) v4i* gas_v4i_ptr;  // global
typedef __attribute__((address_space(3))) v4i* las_v4i_ptr;  // LDS

// ---------------------------------------------------------------------------
// CDNA5 async global->LDS copies (ASYNCcnt path), guarded so compile never
// regresses if the builtin is absent on this toolchain.
// ---------------------------------------------------------------------------
#if defined(__gfx1250__) && __has_builtin(__builtin_amdgcn_global_load_async_to_lds_b128)
#define ASYNC_LDS 1
#else
#define ASYNC_LDS 0
#endif

__device__ __forceinline__ void cp16_async(float* lds_dst, const float* gsrc) {
#if ASYNC_LDS
    __builtin_amdgcn_global_load_async_to_lds_b128(
        (gas_v4i_ptr)gsrc,
        (las_v4i_ptr)lds_dst,
        /*offset=*/0, /*cpol=*/0);
#else
    *(float4*)lds_dst = *(const float4*)gsrc;
#endif
}

__device__ __forceinline__ void async_wait0() {
#if ASYNC_LDS
#if __has_builtin(__builtin_amdgcn_s_wait_asynccnt)
    __builtin_amdgcn_s_wait_asynccnt(0);
#else
    asm volatile("s_wait_asynccnt 0" ::: "memory");
#endif
#endif
}

__device__ __forceinline__ v8f vzero8() {
    v8f z;
#pragma unroll
    for (int i = 0; i < 8; ++i) z[i] = 0.0f;
    return z;
}

// D = A(16x4) * B(4x16) + C, all f32, wave32 WMMA
__device__ __forceinline__ v8f wmma4(v2f a, v2f b, v8f c) {
    return __builtin_amdgcn_wmma_f32_16x16x4_f32(
        /*neg_a=*/false, a, /*neg_b=*/false, b,
        /*c_mod=*/(short)0, c, /*reuse_a=*/false, /*reuse_b=*/false);
}

// A fragment: 16x4 tile at (r0, k0) of row-major [ld] matrix.
__device__ __forceinline__ v2f load_afrag(const float* m, int ld, int r0, int k0, int lane) {
    int mm = lane & 15;
    int kk = (lane >> 4) << 1;
    return *(const v2f*)(m + (r0 + mm) * ld + k0 + kk);
}

// B fragment: 4x16 tile at (k0, c0) of row-major [ld] matrix.
__device__ __forceinline__ v2f load_bfrag(const float* m, int ld, int k0, int c0, int lane) {
    int nn = lane & 15;
    int kk = (lane >> 4) << 1;
    v2f b;
    b.x = m[(k0 + kk)     * ld + c0 + nn];
    b.y = m[(k0 + kk + 1) * ld + c0 + nn];
    return b;
}

// B fragment from a TRANSPOSED source: B[k][n] = m[c0+n][k0+k] (for Q*K^T)
__device__ __forceinline__ v2f load_bfrag_t(const float* m, int ld, int k0, int c0, int lane) {
    int nn = lane & 15;
    int kk = (lane >> 4) << 1;
    return *(const v2f*)(m + (c0 + nn) * ld + k0 + kk);
}

// Cooperative issue of one (64x64 A-panel, 64x64 B-panel) pair.
// A-panel element (r, c) <- Xbase[r*ldX + kb + c]
// B-panel element (r, c) <- Wbase[(kb+r)*ldW + col + c]
__device__ __forceinline__ void issue_panels(const float* Xbase, int ldX,
                                             const float* Wbase, int ldW, int col,
                                             int kb, int tid, float* pX, float* pW) {
#pragma unroll
    for (int t = 0; t < 4; ++t) {
        int idx = tid + t * 256;           // 0..1023 float4 slots
        int r   = idx >> 4;
        int c4  = (idx & 15) << 2;
        cp16_async(&pX[r * 64 + c4], &Xbase[(size_t)r * ldX + kb + c4]);
        cp16_async(&pW[r * 64 + c4], &Wbase[(size_t)(kb + r) * ldW + col + c4]);
    }
}

// ---------------------------------------------------------------------------
// Kernel 1: fused projection + attention per (b, h, c) slice.
//   grid = B*NH*C = 8192 blocks, 256 threads (8 waves).
//   Double-buffered async K-panels; writes softmax probs to attn_out and
//   O-tiles to fcin (ws).
// ---------------------------------------------------------------------------
__global__ __launch_bounds__(256) void attn_fused_kernel(
    const float* __restrict__ q,  const float* __restrict__ k,  const float* __restrict__ v,
    const float* __restrict__ Wq, const float* __restrict__ bq,
    const float* __restrict__ Wk, const float* __restrict__ bk,
    const float* __restrict__ Wv, const float* __restrict__ bv,
    float* __restrict__ attn_out, float* __restrict__ fcin)
{
    __shared__ float smem[16384];  // 64 KB, phase-reused

    const int tid  = threadIdx.x;
    const int lane = tid & 31;
    const int wave = tid >> 5;                 // 0..7
    const int bid  = blockIdx.x;               // b*64 + h*8 + c
    const int b    = bid >> 6;
    const int h    = (bid >> 3) & 7;
    const int c    = bid & 7;
    const int j0   = h * (C_GRP * DK) + c * DK;   // column slice in D_PROJ

    const float* Xs[3]   = { q + (size_t)b * LEN * D_IN,
                             k + (size_t)b * LEN * D_IN,
                             v + (size_t)b * LEN * D_IN };
    const float* Wmat[3] = { Wq, Wk, Wv };
    const float* bias[3] = { bq, bk, bv };

    float* buf0 = smem;            // ping: X-panel | W-panel
    float* buf1 = smem + 8192;     // pong: X-panel | W-panel

    // Per-wave output tiles: tiles 2w and 2w+1 of the 4x4 16x16-tile grid
    v8f acc[3][2];
#pragma unroll
    for (int m = 0; m < 3; ++m)
#pragma unroll
        for (int t = 0; t < 2; ++t) acc[m][t] = vzero8();

    // ---- Phase 1: projections, double-buffered async panels --------------
    issue_panels(Xs[0], D_IN, Wmat[0], D_PROJ, j0, 0, tid, buf0, buf0 + 4096);

#pragma unroll
    for (int m = 0; m < 3; ++m) {
        for (int kbi = 0; kbi < 8; ++kbi) {          // global iter = m*8+kbi
            int par = kbi & 1;                       // m*8 is even
            float* cur = par ? buf1 : buf0;
            float* nxt = par ? buf0 : buf1;

            async_wait0();
            __syncthreads();

            // issue next panel pair (may belong to next matrix)
            if (kbi < 7) {
                issue_panels(Xs[m], D_IN, Wmat[m], D_PROJ, j0,
                             (kbi + 1) * 64, tid, nxt, nxt + 4096);
            } else if (m < 2) {
                issue_panels(Xs[m + 1], D_IN, Wmat[m + 1], D_PROJ, j0,
                             0, tid, nxt, nxt + 4096);
            }

            const float* pX = cur;
            const float* pW = cur + 4096;
#pragma unroll
            for (int t = 0; t < 2; ++t) {
                int tile = wave * 2 + t;
                int tr = (tile >> 2) * 16, tc = (tile & 3) * 16;
                v8f cc = acc[m][t];
#pragma unroll
                for (int kk = 0; kk < 64; kk += 4) {
                    v2f a  = load_afrag(pX, 64, tr, kk, lane);
                    v2f bb = load_bfrag(pW, 64, kk, tc, lane);
                    cc = wmma4(a, bb, cc);
                }
                acc[m][t] = cc;
            }
        }
    }

    // ---- Store Qp/Kp/Vp (+bias) into LDS, reusing panel space ----
    __syncthreads();
    float* Qs = smem;            // 64x64
    float* Ks = smem + 4096;     // 64x64
    float* Vs = smem + 8192;     // 64x64
    float* Ss = smem + 12288;    // 64x64 scores / probs
    {
        int nn = lane & 15, half = lane >> 4;
#pragma unroll
        for (int m = 0; m < 3; ++m) {
            float* dst = smem + m * 4096;
#pragma unroll
            for (int t = 0; t < 2; ++t) {
                int tile = wave * 2 + t;
                int tr = (tile >> 2) * 16, tc = (tile & 3) * 16;
                float bval = bias[m][j0 + tc + nn];
#pragma unroll
                for (int i = 0; i < 8; ++i)
                    dst[(tr + i + 8 * half) * 64 + tc + nn] = acc[m][t][i] + bval;
            }
        }
    }
    __syncthreads();

    // ---- S = (Q * 0.125) @ K^T ----
    {
        int nn = lane & 15, half = lane >> 4;
#pragma unroll
        for (int t = 0; t < 2; ++t) {
            int tile = wave * 2 + t;
            int tr = (tile >> 2) * 16, tc = (tile & 3) * 16;
            v8f s = vzero8();
#pragma unroll
            for (int kk = 0; kk < 64; kk += 4) {
                v2f a  = load_afrag(Qs, 64, tr, kk, lane);
                v2f bb = load_bfrag_t(Ks, 64, kk, tc, lane);
                s = wmma4(a, bb, s);
            }
#pragma unroll
            for (int i = 0; i < 8; ++i)
                Ss[(tr + i + 8 * half) * 64 + tc + nn] = s[i] * 0.125f;
        }
    }
    __syncthreads();

    // ---- Row softmax (4 threads per row, shuffle reductions) ----
    {
        int row = tid >> 2;
        int seg = (tid & 3) << 4;   // 16 columns per thread
        float e[16];
        float mx = -3.4e38f;
#pragma unroll
        for (int i = 0; i < 16; ++i) {
            e[i] = Ss[row * 64 + seg + i];
            mx = fmaxf(mx, e[i]);
        }
        mx = fmaxf(mx, __shfl_xor(mx, 1, 32));
        mx = fmaxf(mx, __shfl_xor(mx, 2, 32));
        float sum = 0.0f;
#pragma unroll
        for (int i = 0; i < 16; ++i) { e[i] = __expf(e[i] - mx); sum += e[i]; }
        sum += __shfl_xor(sum, 1, 32);
        sum += __shfl_xor(sum, 2, 32);
        float inv = 1.0f / sum;

        // attn layout: [b][h][c][qrow][kcol] -> base = bid*4096
        size_t abase = (size_t)bid * 4096 + (size_t)row * 64 + seg;
#pragma unroll
        for (int g = 0; g < 4; ++g) {
            float4 pv;
            pv.x = e[4 * g + 0] * inv;
            pv.y = e[4 * g + 1] * inv;
            pv.z = e[4 * g + 2] * inv;
            pv.w = e[4 * g + 3] * inv;
            *(float4*)&Ss[row * 64 + seg + 4 * g] = pv;
            *(float4*)&attn_out[abase + 4 * g]    = pv;
        }
    }
    __syncthreads();

    // ---- O = P @ V, store directly as FC-input row (b*64 + c*8 + h) ----
    {
        size_t obase = ((size_t)b * 64 + (size_t)c * 8 + h) * (size_t)D_PROJ;
        int nn = lane & 15, half = lane >> 4;
#pragma unroll
        for (int t = 0; t < 2; ++t) {
            int tile = wave * 2 + t;
            int tr = (tile >> 2) * 16, tc = (tile & 3) * 16;
            v8f o = vzero8();
#pragma unroll
            for (int kk = 0; kk < 64; kk += 4) {
                v2f a  = load_afrag(Ss, 64, tr, kk, lane);
                v2f bb = load_bfrag(Vs, 64, kk, tc, lane);
                o = wmma4(a, bb, o);
            }
#pragma unroll
            for (int i = 0; i < 8; ++i)
                fcin[obase + (size_t)(tr + i + 8 * half) * 64 + tc + nn] = o[i];
        }
    }
}

// ---------------------------------------------------------------------------
// Kernel 2: FC (8192x4096)@(4096x512) + bias + residual(q rows).
//   grid = (128, 8), 256 threads, 64x64 output tile, double-buffered async.
// ---------------------------------------------------------------------------
__global__ __launch_bounds__(256) void fc_residual_kernel(
    const float* __restrict__ fcin, const float* __restrict__ Wfc,
    const float* __restrict__ bfc,  const float* __restrict__ qres,
    float* __restrict__ out)
{
    __shared__ float smem[16384];   // 64 KB: 2 x (A panel + B panel)
    float* buf0 = smem;
    float* buf1 = smem + 8192;

    const int tid  = threadIdx.x;
    const int lane = tid & 31;
    const int wave = tid >> 5;
    const int row0 = blockIdx.x * 64;   // 0..8191
    const int col0 = blockIdx.y * 64;   // 0..511

    const float* Abase = fcin + (size_t)row0 * D_PROJ;

    v8f acc[2];
#pragma unroll
    for (int t = 0; t < 2; ++t) acc[t] = vzero8();

    issue_panels(Abase, D_PROJ, Wfc, D_IN, col0, 0, tid, buf0, buf0 + 4096);

    for (int kbi = 0; kbi < 64; ++kbi) {
        int par = kbi & 1;
        float* cur = par ? buf1 : buf0;
        float* nxt = par ? buf0 : buf1;

        async_wait0();
        __syncthreads();

        if (kbi < 63)
            issue_panels(Abase, D_PROJ, Wfc, D_IN, col0,
                         (kbi + 1) * 64, tid, nxt, nxt + 4096);

        const float* pA = cur;
        const float* pB = cur + 4096;
#pragma unroll
        for (int t = 0; t < 2; ++t) {
            int tile = wave * 2 + t;
            int tr = (tile >> 2) * 16, tc = (tile & 3) * 16;
            v8f cc = acc[t];
#pragma unroll
            for (int kk = 0; kk < 64; kk += 4) {
                v2f a  = load_afrag(pA, 64, tr, kk, lane);
                v2f bb = load_bfrag(pB, 64, kk, tc, lane);
                cc = wmma4(a, bb, cc);
            }
            acc[t] = cc;
        }
    }

    int nn = lane & 15, half = lane >> 4;
#pragma unroll
    for (int t = 0; t < 2; ++t) {
        int tile = wave * 2 + t;
        int tr = (tile >> 2) * 16, tc = (tile & 3) * 16;
        float bval = bfc[col0 + tc + nn];
#pragma unroll
        for (int i = 0; i < 8; ++i) {
            size_t row = (size_t)(row0 + tr + i + 8 * half);
            int    col = col0 + tc + nn;
            out[row * D_IN + col] = acc[t][i] + bval + qres[row * D_IN + col];
        }
    }
}

extern "C" void kernel_launch(void* const* d_in, const int* in_sizes, int n_in,
                              void* d_out, int out_size, void* d_ws, size_t ws_size,
                              hipStream_t stream)
{
    (void)in_sizes; (void)n_in; (void)out_size; (void)ws_size;

    const float* q   = (const float*)d_in[0];
    const float* k   = (const float*)d_in[1];
    const float* v   = (const float*)d_in[2];
    const float* Wq  = (const float*)d_in[3];
    const float* bq  = (const float*)d_in[4];
    const float* Wk  = (const float*)d_in[5];
    const float* bk  = (const float*)d_in[6];
    const float* Wv  = (const float*)d_in[7];
    const float* bv  = (const float*)d_in[8];
    const float* Wfc = (const float*)d_in[9];
    const float* bfc = (const float*)d_in[10];

    float* out  = (float*)d_out;                                  // (B, LEN, 512)
    float* attn = out + (size_t)B_SZ * LEN * D_IN;                // (B, NH, C, 64, 64)
    float* fcin = (float*)d_ws;                                   // (8192, 4096) staging

    attn_fused_kernel<<<dim3(B_SZ * NHEAD * C_GRP), dim3(256), 0, stream>>>(
        q, k, v, Wq, bq, Wk, bk, Wv, bv, attn, fcin);

    fc_residual_kernel<<<dim3((B_SZ * LEN) / 64, D_IN / 64), dim3(256), 0, stream>>>(
        fcin, Wfc, bfc, q, out);
}